// KGAT_smodal_58067957842413
// MI455X (gfx1250) — compile-verified
//
#include <hip/hip_runtime.h>
#include <hip/hip_bf16.h>
#include <math.h>

#define D_FEAT 64
#define K_NEI 32
#define NEG_SLOPE 0.2f

typedef float v2f __attribute__((ext_vector_type(2)));
typedef float v8f __attribute__((ext_vector_type(8)));

// ---------------------------------------------------------------------------
// K1: H[N x 64] = A[N x 64] * W[64 x 64] using V_WMMA_F32_16X16X4_F32.
// One wave per 16x16 output tile; 4 waves per block cover the 4 column tiles.
// A-frag (16x4 f32): lanes 0-15 -> K = k0+{0,1}, lanes 16-31 -> K = k0+{2,3},
// row M = lane%16.  B-frag (4x16 f32) mirrored: VGPR0 holds rows {k0,k0+2},
// VGPR1 holds rows {k0+1,k0+3}, col N = lane%16.
// C/D: VGPR v -> M = v + 8*(lane/16), N = lane%16.
// ---------------------------------------------------------------------------
__global__ void gemm_h_kernel(const float* __restrict__ A,
                              const float* __restrict__ W,
                              float* __restrict__ H, int n) {
  const int wave = threadIdx.x >> 5;   // 0..3 : column tile
  const int lane = threadIdx.x & 31;
  const int row0 = blockIdx.x * 16;
  if (row0 + 16 > n) return;           // uniform per block (EXEC stays all-1s)
  const int col0 = wave * 16;
  const int half = lane >> 4;          // 0 = lanes 0-15, 1 = lanes 16-31
  const int l    = lane & 15;

  v8f c = {};
  const float* arow = A + (size_t)(row0 + l) * D_FEAT;
#pragma unroll
  for (int k0 = 0; k0 < D_FEAT; k0 += 4) {
    const int ka = k0 + 2 * half;
    v2f a, b;
    a.x = arow[ka + 0];
    a.y = arow[ka + 1];
    b.x = W[(ka + 0) * D_FEAT + col0 + l];
    b.y = W[(ka + 1) * D_FEAT + col0 + l];
    // 8 args: (neg_a, A, neg_b, B, c_mod, C, reuse_a, reuse_b)
    c = __builtin_amdgcn_wmma_f32_16x16x4_f32(false, a, false, b, (short)0, c,
                                              false, false);
  }
#pragma unroll
  for (int v = 0; v < 8; ++v) {
    H[(size_t)(row0 + v + 8 * half) * D_FEAT + col0 + l] = c[v];
  }
}

// ---------------------------------------------------------------------------
// K2: s_i[r] = h[r] . att[0:64],  s_j[r] = h[r] . att[64:128]
// Wave per row; lane covers dims {lane, lane+32}; wave32 shfl_xor reduction.
// ---------------------------------------------------------------------------
__global__ void sij_kernel(const float* __restrict__ H,
                           const float* __restrict__ att,
                           float* __restrict__ s_i, float* __restrict__ s_j,
                           int n) {
  const int wid  = (blockIdx.x * blockDim.x + threadIdx.x) >> 5;
  const int lane = threadIdx.x & 31;
  if (wid >= n) return;
  const float* hr = H + (size_t)wid * D_FEAT;
  const float h0 = hr[lane];
  const float h1 = hr[lane + 32];
  float pi = h0 * att[lane] + h1 * att[lane + 32];
  float pj = h0 * att[D_FEAT + lane] + h1 * att[D_FEAT + lane + 32];
#pragma unroll
  for (int off = 16; off > 0; off >>= 1) {
    pi += __shfl_xor(pi, off, 32);
    pj += __shfl_xor(pj, off, 32);
  }
  if (lane == 0) { s_i[wid] = pi; s_j[wid] = pj; }
}

// ---------------------------------------------------------------------------
// K3: init scratch + output accumulator (harness does not re-zero between runs)
// ---------------------------------------------------------------------------
__global__ void init_kernel(float* __restrict__ out, float* __restrict__ m,
                            float* __restrict__ denom, int n) {
  const int t = blockIdx.x * blockDim.x + threadIdx.x;
  if (t < n * D_FEAT) out[t] = 0.0f;
  if (t < n) {
    m[t]     = __int_as_float(0xff800000);  // -inf
    denom[t] = 0.0f;
  }
}

// Portable float atomic-max via ordered-int bit tricks (init must be -inf).
__device__ inline void atomicMaxF(float* addr, float val) {
  if (val >= 0.0f)
    atomicMax((int*)addr, __float_as_int(val));
  else
    atomicMin((unsigned int*)addr, __float_as_uint(val));
}

// Decode edge id t in [0, n*K + n): first n*K entries come from edges_matrix
// (src = t/K, dst = edges[t]), last n are self-loops. Returns false for the
// invalid (src==dst) entries, whose exp() contribution is exactly 0.
__device__ inline bool edge_decode(int t, int n, const int* __restrict__ edges,
                                   int& src, int& dst) {
  const int E = n * K_NEI;
  if (t < E) {
    src = t / K_NEI;          // K_NEI = 32 -> shift
    dst = edges[t];
    return dst != src;
  }
  src = dst = t - E;
  return true;
}

__device__ inline float leaky(float a) {
  return (a >= 0.0f) ? a : NEG_SLOPE * a;
}

// ---------------------------------------------------------------------------
// K4: segment max of alpha over dst
// ---------------------------------------------------------------------------
__global__ void alpha_max_kernel(const int* __restrict__ edges,
                                 const float* __restrict__ s_i,
                                 const float* __restrict__ s_j,
                                 float* __restrict__ m, int n) {
  const int t = blockIdx.x * blockDim.x + threadIdx.x;
  if (t >= n * (K_NEI + 1)) return;
  int src, dst;
  if (!edge_decode(t, n, edges, src, dst)) return;
  atomicMaxF(&m[dst], leaky(s_i[dst] + s_j[src]));
}

// ---------------------------------------------------------------------------
// K5: denom[dst] = sum exp(alpha - m[dst])
// ---------------------------------------------------------------------------
__global__ void denom_kernel(const int* __restrict__ edges,
                             const float* __restrict__ s_i,
                             const float* __restrict__ s_j,
                             const float* __restrict__ m,
                             float* __restrict__ denom, int n) {
  const int t = blockIdx.x * blockDim.x + threadIdx.x;
  if (t >= n * (K_NEI + 1)) return;
  int src, dst;
  if (!edge_decode(t, n, edges, src, dst)) return;
  const float a = leaky(s_i[dst] + s_j[src]);
  atomicAdd(&denom[dst], expf(a - m[dst]));
}

// ---------------------------------------------------------------------------
// K6: out[dst] += h[src] * softmax_weight.  Wave per edge; lane owns dims
// {lane, lane+32} -> two coalesced 128B gathers + two coalesced atomic bursts.
// h and out are L2-resident (25.6 MB each << 192 MB).
// ---------------------------------------------------------------------------
__global__ void aggregate_kernel(const int* __restrict__ edges,
                                 const float* __restrict__ s_i,
                                 const float* __restrict__ s_j,
                                 const float* __restrict__ m,
                                 const float* __restrict__ denom,
                                 const float* __restrict__ H,
                                 float* __restrict__ out, int n) {
  const int wid  = (blockIdx.x * blockDim.x + threadIdx.x) >> 5;
  const int lane = threadIdx.x & 31;
  if (wid >= n * (K_NEI + 1)) return;
  int src, dst;
  if (!edge_decode(wid, n, edges, src, dst)) return;
  const float al = leaky(s_i[dst] + s_j[src]);
  const float a  = expf(al - m[dst]) / (denom[dst] + 1e-16f);
  const float v0 = H[(size_t)src * D_FEAT + lane]      * a;
  const float v1 = H[(size_t)src * D_FEAT + lane + 32] * a;
  atomicAdd(&out[(size_t)dst * D_FEAT + lane],      v0);
  atomicAdd(&out[(size_t)dst * D_FEAT + lane + 32], v1);
}

// ---------------------------------------------------------------------------
// K7: out = (out + bias) / max(||out + bias||, 1e-12), in place
// ---------------------------------------------------------------------------
__global__ void finalize_kernel(float* __restrict__ out,
                                const float* __restrict__ bias, int n) {
  const int wid  = (blockIdx.x * blockDim.x + threadIdx.x) >> 5;
  const int lane = threadIdx.x & 31;
  if (wid >= n) return;
  float x0 = out[(size_t)wid * D_FEAT + lane]      + bias[lane];
  float x1 = out[(size_t)wid * D_FEAT + lane + 32] + bias[lane + 32];
  float ss = x0 * x0 + x1 * x1;
#pragma unroll
  for (int off = 16; off > 0; off >>= 1) ss += __shfl_xor(ss, off, 32);
  const float inv = 1.0f / fmaxf(sqrtf(ss), 1e-12f);
  out[(size_t)wid * D_FEAT + lane]      = x0 * inv;
  out[(size_t)wid * D_FEAT + lane + 32] = x1 * inv;
}

// ---------------------------------------------------------------------------
extern "C" void kernel_launch(void* const* d_in, const int* in_sizes, int n_in,
                              void* d_out, int out_size, void* d_ws,
                              size_t ws_size, hipStream_t stream) {
  const float* all_embed = (const float*)d_in[0];  // [N, 64]
  const float* W         = (const float*)d_in[1];  // [64, 64]
  const float* att       = (const float*)d_in[2];  // [1, 1, 128]
  const float* bias      = (const float*)d_in[3];  // [64]
  const int*   edges     = (const int*)d_in[7];    // [N, 32]
  float*       out       = (float*)d_out;          // [N, 64]

  const int n = in_sizes[0] / D_FEAT;              // N_NODES

  // Workspace layout (floats): h[N*64] | s_i[N] | s_j[N] | m[N] | denom[N]
  float* h     = (float*)d_ws;
  float* s_i   = h + (size_t)n * D_FEAT;
  float* s_j   = s_i + n;
  float* m     = s_j + n;
  float* denom = m + n;

  const int T = n * (K_NEI + 1);                   // total edges incl. loops

  // K1: GEMM (wave per 16x16 tile, 4 waves/block)
  gemm_h_kernel<<<(n + 15) / 16, 128, 0, stream>>>(all_embed, W, h, n);
  // K2: attention pre-scores
  sij_kernel<<<(n * 32 + 255) / 256, 256, 0, stream>>>(h, att, s_i, s_j, n);
  // K3: init accumulators
  init_kernel<<<(n * D_FEAT + 255) / 256, 256, 0, stream>>>(out, m, denom, n);
  // K4: segment max
  alpha_max_kernel<<<(T + 255) / 256, 256, 0, stream>>>(edges, s_i, s_j, m, n);
  // K5: softmax denominator
  denom_kernel<<<(T + 255) / 256, 256, 0, stream>>>(edges, s_i, s_j, m, denom, n);
  // K6: weighted scatter aggregation (wave per edge)
  {
    const long long threads = (long long)T * 32;
    aggregate_kernel<<<(int)((threads + 255) / 256), 256, 0, stream>>>(
        edges, s_i, s_j, m, denom, h, out, n);
  }
  // K7: bias + L2 normalize
  finalize_kernel<<<(n * 32 + 255) / 256, 256, 0, stream>>>(out, bias, n);
}